// Matcher_41214506172832
// MI455X (gfx1250) — compile-verified
//
#include <hip/hip_runtime.h>
#include <hip/hip_bf16.h>

// 32 independent 128x128 Hungarian problems (B*T = 4*8). One wave32 per
// problem: all inner-loop coordination via wave shuffles, cost matrix in LDS,
// cost built with v_wmma_f32_16x16x4_f32 (rank-2 broadcast-subtract).
// u-dual scatter updates are deferred to registers and flushed once per row,
// so the latency-critical inner loop does exactly 2 DS loads per iteration.

typedef float v2f __attribute__((ext_vector_type(2)));
typedef float v8f __attribute__((ext_vector_type(8)));

static constexpr int NB   = 32;    // B*T problems
static constexpr int NN   = 128;   // square size (Q == N)
static constexpr float BIGF = 1e9f;

struct Smem {
    float cost[NN * NN];           // 64 KB: cost[i*128 + j]
    float px[NN], py[NN], tx[NN], ty[NN];
    float u[NN + 4];               // row duals, indexed 1..128
};

__device__ __forceinline__ int geti4(const int a[4], int k) {
    return k == 0 ? a[0] : k == 1 ? a[1] : k == 2 ? a[2] : a[3];
}
__device__ __forceinline__ void seti4(int a[4], int k, int v) {
    if (k == 0) a[0] = v; else if (k == 1) a[1] = v; else if (k == 2) a[2] = v; else a[3] = v;
}

__global__ __launch_bounds__(32) void hungarian_kernel(const float* __restrict__ preds,
                                                       const float* __restrict__ targets,
                                                       int* __restrict__ out) {
    __shared__ Smem s;
    const int bt   = blockIdx.x;
    const int lane = threadIdx.x;          // 0..31 (wave32)

    // ---- stage points + zero duals ----------------------------------------
    const float* P = preds   + bt * NN * 2;
    const float* T = targets + bt * NN * 2;
    #pragma unroll
    for (int k = 0; k < 4; ++k) {
        int idx = lane * 4 + k;
        s.px[idx] = P[2 * idx];
        s.py[idx] = P[2 * idx + 1];
        s.tx[idx] = T[2 * idx];
        s.ty[idx] = T[2 * idx + 1];
        s.u[k * 32 + lane] = 0.0f;
    }
    if (lane == 0) s.u[NN] = 0.0f;
    __syncthreads();   // single-wave WG: lowers to waits + s_nop

    // ---- cost matrix via WMMA f32 16x16x4 ---------------------------------
    // dx = A x B, A(16x4) = [px, 1, 0, 0], B(4x16) = [1; -tx; 0; 0]
    // A layout (ISA 16x4 f32 table): lanes0-15 VGPR{0,1}=K{0,1}; lanes16-31 K{2,3}
    // B layout (K split across lane halves): lanes0-15 VGPR{0,1}=row K{0,1}; upper half = 0
    // Unconditional LDS loads + multiplicative mask (exact for m in {0,1})
    // avoid exec-mask save/restore around every load.
    const int   half = lane >> 4;          // 0 or 1
    const int   l16  = lane & 15;
    const float m    = half ? 0.0f : 1.0f; // zero out K=2,3 contributions
    const v8f czero = {0.f,0.f,0.f,0.f,0.f,0.f,0.f,0.f};
    for (int ti = 0; ti < 8; ++ti) {
        const float pxv = s.px[ti * 16 + l16];
        const float pyv = s.py[ti * 16 + l16];
        v2f ax, ay;
        ax.x = m * pxv;  ax.y = m;
        ay.x = m * pyv;  ay.y = m;
        for (int tj = 0; tj < 8; ++tj) {
            const float txv = s.tx[tj * 16 + l16];
            const float tyv = s.ty[tj * 16 + l16];
            v2f bx, by;
            bx.x = m;  bx.y = m * (-txv);
            by.x = m;  by.y = m * (-tyv);
            v8f dx = __builtin_amdgcn_wmma_f32_16x16x4_f32(false, ax, false, bx,
                                                           (short)0, czero, false, false);
            v8f dy = __builtin_amdgcn_wmma_f32_16x16x4_f32(false, ay, false, by,
                                                           (short)0, czero, false, false);
            // D layout: VGPR r -> row M=r (lanes0-15) / M=r+8 (lanes16-31), N=lane%16
            const int rowb = ti * 16 + half * 8;
            const int col  = tj * 16 + l16;
            #pragma unroll
            for (int r = 0; r < 8; ++r) {
                s.cost[(rowb + r) * NN + col] = fabsf(dx[r]) + fabsf(dy[r]);
            }
        }
    }
    __syncthreads();

    // ---- Hungarian (e-maxx shortest augmenting path), lane owns cols ------
    // column j (1..128): owner lane = (j-1)>>2, slot k = (j-1)&3
    float vv[4]   = {0.f, 0.f, 0.f, 0.f};  // v[j]
    int   p_[4]   = {0, 0, 0, 0};          // p[j]: 1-indexed row matched to col j
    float minv[4];
    int   way[4];
    bool  used[4];
    float du_[4];                          // deferred u[p[j]] accumulation

    for (int i = 0; i < NN; ++i) {
        const int p0 = i + 1;              // p[0] = i+1
        #pragma unroll
        for (int k = 0; k < 4; ++k) {
            minv[k] = BIGF; way[k] = 0; used[k] = false; du_[k] = 0.0f;
        }
        float duSum = 0.0f;                // deferred u[p0] accumulation
        int j0  = 0;                       // scalar (uniform)
        int pj0 = p0;                      // p[j0], maintained across iterations

        for (;;) {
            // mark used[j0] (skipped for virtual column 0)
            if (j0 != 0) {
                int owner = (j0 - 1) >> 2, kk = (j0 - 1) & 3;
                if (lane == owner) {
                    if (kk == 0) used[0] = true; else if (kk == 1) used[1] = true;
                    else if (kk == 2) used[2] = true; else used[3] = true;
                }
            }
            const int   i0   = pj0;        // row on path (1..128), uniform scalar
            const float u_i0 = s.u[i0];    // uniform broadcast LDS read
                                           // (never has pending deferred adds: each
                                           // row's u is read once, when first used)

            // reduced costs for my 4 contiguous columns: one ds_load_b128
            const float4 c4 = *reinterpret_cast<const float4*>(&s.cost[(i0 - 1) * NN + lane * 4]);
            const float cc[4] = {c4.x, c4.y, c4.z, c4.w};
            #pragma unroll
            for (int k = 0; k < 4; ++k) {
                if (!used[k]) {
                    float cur = (cc[k] - u_i0) - vv[k];
                    if (cur < minv[k]) { minv[k] = cur; way[k] = j0; }
                }
            }

            // argmin over masked minv (first-minimum tie-break, like jnp.argmin)
            float bv = used[0] ? BIGF : minv[0];
            int   bi = lane * 4 + 1;
            #pragma unroll
            for (int k = 1; k < 4; ++k) {
                float mk = used[k] ? BIGF : minv[k];
                if (mk < bv) { bv = mk; bi = lane * 4 + k + 1; }
            }
            #pragma unroll
            for (int off = 16; off >= 1; off >>= 1) {
                float ov = __shfl_xor(bv, off, 32);
                int   oi = __shfl_xor(bi, off, 32);
                if (ov < bv || (ov == bv && oi < bi)) { bv = ov; bi = oi; }
            }
            const float delta = bv;                                   // uniform
            const int   j1    = __builtin_amdgcn_readfirstlane(bi);   // scalar

            // dual updates: defer u scatter-adds to registers (flush per row)
            duSum += delta;
            #pragma unroll
            for (int k = 0; k < 4; ++k) {
                if (used[k]) { du_[k] += delta; vv[k] -= delta; }
                else         { minv[k] -= delta; }
            }

            j0 = j1;
            {   // pj0 = p[j0] via shuffle from owner, scalarized
                int owner = (j0 - 1) >> 2, kk = (j0 - 1) & 3;
                pj0 = __builtin_amdgcn_readfirstlane(__shfl(geti4(p_, kk), owner, 32));
            }
            if (pj0 == 0) break;           // reached a free column
        }

        // flush deferred u updates (distinct rows; must use pre-augment p)
        #pragma unroll
        for (int k = 0; k < 4; ++k) {
            if (used[k]) s.u[p_[k]] += du_[k];
        }
        if (lane == 0) s.u[p0] += duSum;

        // augment along `way` chain back to column 0 (uniform serial walk)
        int j = j0;
        while (j != 0) {
            int owner = (j - 1) >> 2, kk = (j - 1) & 3;
            int j1 = __builtin_amdgcn_readfirstlane(__shfl(geti4(way, kk), owner, 32));
            int pv;
            if (j1 == 0) {
                pv = p0;
            } else {
                int o2 = (j1 - 1) >> 2, k2 = (j1 - 1) & 3;
                pv = __builtin_amdgcn_readfirstlane(__shfl(geti4(p_, k2), o2, 32));
            }
            if (lane == owner) seti4(p_, kk, pv);       // p[j] = p[j1]
            j = j1;
        }
    }

    // ---- emit row_to_col: out[p[j]-1] = j-1 (perfect matching covers all) --
    int* o = out + bt * NN;
    #pragma unroll
    for (int k = 0; k < 4; ++k) {
        o[p_[k] - 1] = lane * 4 + k;
    }
}

extern "C" void kernel_launch(void* const* d_in, const int* in_sizes, int n_in,
                              void* d_out, int out_size, void* d_ws, size_t ws_size,
                              hipStream_t stream) {
    (void)in_sizes; (void)n_in; (void)d_ws; (void)ws_size; (void)out_size;
    const float* preds   = (const float*)d_in[0];   // (4,8,128,2) f32
    const float* targets = (const float*)d_in[1];   // (4,8,128,2) f32
    int* out = (int*)d_out;                         // (4,8,128) int32
    hipLaunchKernelGGL(hungarian_kernel, dim3(NB), dim3(32), 0, stream,
                       preds, targets, out);
}